// LINK_23596550324873
// MI455X (gfx1250) — compile-verified
//
#include <hip/hip_runtime.h>
#include <stdint.h>

// ---------------------------------------------------------------------------
// Sparse scatter-add:  out[r,:] = b + sum_{edges (r,c)} Wt[c,:]
// N=100000 nodes, C=64 channels, E=3.2M edges.
// Memory/atomic bound (205 MFLOP total) -> optimize data movement:
//   * transpose W once so gathers are contiguous 256B rows (L2-resident)
//   * double-buffered async-to-LDS staging of the edge stream (CDNA5 path)
//   * half-wave-per-edge: 16 lanes x float4 coalesced gather + 4 f32 atomics
// ---------------------------------------------------------------------------

#define TILE_EDGES    1024
#define SCATTER_BLOCK 256
#define SCATTER_GRID  1024
#define RED_GRID      1024
#define RED_BLOCK     256
#define IMAX_I32      0x7FFFFFFF

#if __has_builtin(__builtin_amdgcn_global_load_async_to_lds_b128)
#define USE_ASYNC_LDS 1
#else
#define USE_ASYNC_LDS 0
#endif

typedef int v4i __attribute__((ext_vector_type(4)));
typedef __attribute__((address_space(1))) v4i* as1_v4i_p;
typedef __attribute__((address_space(3))) v4i* as3_v4i_p;

__device__ __forceinline__ void async_load_b128(const void* gsrc, void* lds_dst) {
#if USE_ASYNC_LDS
  // generic->AS1 via 64-bit integer (global generic addr == AS1 addr);
  // generic->AS3 via low 32 bits (flat LDS aperture: ADDR[31:0] = LDS offset).
  as1_v4i_p g = (as1_v4i_p)(unsigned long long)(uintptr_t)gsrc;
  as3_v4i_p l = (as3_v4i_p)(unsigned int)(uintptr_t)lds_dst;
  __builtin_amdgcn_global_load_async_to_lds_b128(g, l, 0, 0);
#else
  *(int4*)lds_dst = *(const int4*)gsrc;   // synchronous fallback
#endif
}

__device__ __forceinline__ void wait_async_all() {
#if USE_ASYNC_LDS
#if __has_builtin(__builtin_amdgcn_s_wait_asynccnt)
  __builtin_amdgcn_s_wait_asynccnt(0);
#else
  asm volatile("s_wait_asynccnt 0" ::: "memory");
#endif
#endif
}

// ---------------------------------------------------------------------------
__global__ void k_init_rmin(int* rmin) {
  if (threadIdx.x == 0 && blockIdx.x == 0) *rmin = IMAX_I32;
}

__global__ __launch_bounds__(RED_BLOCK)
void k_row_min(const int* __restrict__ rows, int n_edges, int* __restrict__ rmin) {
  int v = IMAX_I32;
  for (int i = blockIdx.x * blockDim.x + threadIdx.x; i < n_edges;
       i += gridDim.x * blockDim.x)
    v = min(v, rows[i]);
  // wave32 reduction
  #pragma unroll
  for (int off = 16; off > 0; off >>= 1)
    v = min(v, __shfl_xor(v, off, 32));
  __shared__ int s[RED_BLOCK / 32];
  const int lane = threadIdx.x & 31, wid = threadIdx.x >> 5;
  if (lane == 0) s[wid] = v;
  __syncthreads();
  if (threadIdx.x == 0) {
    int bm = s[0];
    #pragma unroll
    for (int i = 1; i < RED_BLOCK / 32; ++i) bm = min(bm, s[i]);
    atomicMin(rmin, bm);
  }
}

// out[i] = b[i % 64], vectorized float4
__global__ __launch_bounds__(256)
void k_init_out(float* __restrict__ out, const float* __restrict__ b, int total4) {
  __shared__ float sb[64];
  if (threadIdx.x < 64) sb[threadIdx.x] = b[threadIdx.x];
  __syncthreads();
  for (int i = blockIdx.x * blockDim.x + threadIdx.x; i < total4;
       i += gridDim.x * blockDim.x) {
    const int c = (i * 4) & 63;
    const float4 v = make_float4(sb[c], sb[c + 1], sb[c + 2], sb[c + 3]);
    reinterpret_cast<float4*>(out)[i] = v;
  }
}

// W[64][n] -> Wt[n][64], 32x32 LDS tiles, both sides coalesced
__global__ __launch_bounds__(256)
void k_transpose(const float* __restrict__ W, float* __restrict__ Wt, int n) {
  __shared__ float tile[32][33];
  const int nBase = blockIdx.x * 32;
  const int cBase = blockIdx.y * 32;
  const int x = threadIdx.x;   // 0..31
  const int y0 = threadIdx.y;  // 0..7
  #pragma unroll
  for (int j = 0; j < 32; j += 8) {
    const int c  = cBase + y0 + j;
    const int nn = nBase + x;
    tile[y0 + j][x] = (nn < n) ? W[(size_t)c * (size_t)n + nn] : 0.0f;
  }
  __syncthreads();
  #pragma unroll
  for (int j = 0; j < 32; j += 8) {
    const int nn = nBase + y0 + j;
    if (nn < n) Wt[(size_t)nn * 64 + cBase + x] = tile[x][y0 + j];
  }
}

// ---------------------------------------------------------------------------
__device__ __forceinline__ void scatter_one_edge(
    int row, int col, int c4, const float* __restrict__ Wsrc, float* __restrict__ out,
    int rmin, int n_nodes, int transposed) {
  float4 wv;
  if (transposed) {
    wv = *reinterpret_cast<const float4*>(Wsrc + (size_t)col * 64 + c4);
  } else {
    const float* base = Wsrc + (size_t)c4 * (size_t)n_nodes + col;
    wv.x = base[0];
    wv.y = base[(size_t)n_nodes];
    wv.z = base[2 * (size_t)n_nodes];
    wv.w = base[3 * (size_t)n_nodes];
  }
  float* o = out + (size_t)(row - rmin) * 64 + c4;
  unsafeAtomicAdd(o + 0, wv.x);   // global_atomic_add_f32
  unsafeAtomicAdd(o + 1, wv.y);
  unsafeAtomicAdd(o + 2, wv.z);
  unsafeAtomicAdd(o + 3, wv.w);
}

__global__ __launch_bounds__(SCATTER_BLOCK)
void k_scatter(const int* __restrict__ rows, const int* __restrict__ cols,
               const float* __restrict__ Wsrc, float* __restrict__ out,
               const int* __restrict__ rmin_p, int num_tiles, int n_nodes,
               int transposed) {
  __shared__ int s_rows[2][TILE_EDGES];
  __shared__ int s_cols[2][TILE_EDGES];
  const int rmin = *rmin_p;                 // scalar (s_load)
  const int tid  = threadIdx.x;
  const int lane = tid & 31;
  const int wid  = tid >> 5;
  const int half = lane >> 4;               // which edge of the lane-pair
  const int c4   = (lane & 15) * 4;         // channel base: 16 lanes x float4 = 64ch

  int t = blockIdx.x;
  if (t >= num_tiles) return;

  // stage first tile (buffer 0) asynchronously
  async_load_b128(rows + (size_t)t * TILE_EDGES + tid * 4, &s_rows[0][tid * 4]);
  async_load_b128(cols + (size_t)t * TILE_EDGES + tid * 4, &s_cols[0][tid * 4]);

  int buf = 0;
  for (; t < num_tiles; t += gridDim.x) {
    wait_async_all();       // my staged tile (buf) landed in LDS
    __syncthreads();        // everyone's staged tile landed

    const int tn = t + gridDim.x;
    if (tn < num_tiles) {   // prefetch next tile into the other buffer
      async_load_b128(rows + (size_t)tn * TILE_EDGES + tid * 4, &s_rows[buf ^ 1][tid * 4]);
      async_load_b128(cols + (size_t)tn * TILE_EDGES + tid * 4, &s_cols[buf ^ 1][tid * 4]);
    }

    const int ebase = wid * (TILE_EDGES / 8);  // 128 edges per wave
    #pragma unroll 4
    for (int i = 0; i < TILE_EDGES / 8; i += 2) {
      const int e   = ebase + i + half;        // 2 edges per wave-iteration
      const int row = s_rows[buf][e];          // LDS broadcast (2 groups)
      const int col = s_cols[buf][e];
      scatter_one_edge(row, col, c4, Wsrc, out, rmin, n_nodes, transposed);
    }
    __syncthreads();        // all reads of buf done before it is restaged
    buf ^= 1;
  }
}

__global__ __launch_bounds__(256)
void k_scatter_tail(const int* __restrict__ rows, const int* __restrict__ cols,
                    const float* __restrict__ Wsrc, float* __restrict__ out,
                    const int* __restrict__ rmin_p, int e0, int n_edges,
                    int n_nodes, int transposed) {
  const int rmin = *rmin_p;
  const int lane = threadIdx.x & 31;
  const int half = lane >> 4;
  const int c4   = (lane & 15) * 4;
  const int gw   = (blockIdx.x * blockDim.x + threadIdx.x) >> 5;
  const int e    = e0 + gw * 2 + half;
  if (e >= n_edges) return;
  scatter_one_edge(rows[e], cols[e], c4, Wsrc, out, rmin, n_nodes, transposed);
}

// ---------------------------------------------------------------------------
extern "C" void kernel_launch(void* const* d_in, const int* in_sizes, int n_in,
                              void* d_out, int out_size, void* d_ws, size_t ws_size,
                              hipStream_t stream) {
  const int*   edge = (const int*)d_in[0];    // [2][E]
  const float* W    = (const float*)d_in[1];  // [C][N]
  const float* b    = (const float*)d_in[2];  // [C]
  float*       out  = (float*)d_out;          // [N][C]

  const int E  = in_sizes[0] / 2;
  const int C  = in_sizes[2];                 // 64
  const int Nn = in_sizes[1] / (C > 0 ? C : 1);

  const int* rows = edge;
  const int* cols = edge + E;

  // workspace layout: [0..3] row-min, [256..] Wt (N x 64 f32)
  int*         rmin     = (int*)d_ws;
  const size_t wt_off   = 256;
  const size_t wt_bytes = (size_t)Nn * 64 * sizeof(float);
  const int    use_wt   = (ws_size >= wt_off + wt_bytes) ? 1 : 0;  // constant per run
  float*       Wt       = (float*)((char*)d_ws + wt_off);

  // 1) row minimum (faithful to reference's row -= row.min())
  k_init_rmin<<<1, 1, 0, stream>>>(rmin);
  k_row_min<<<RED_GRID, RED_BLOCK, 0, stream>>>(rows, E, rmin);

  // 2) out = broadcast bias
  const int total4 = (Nn * C) / 4;
  k_init_out<<<1024, 256, 0, stream>>>(out, b, total4);

  // 3) transpose W -> Wt so each gather is one contiguous 256B row
  if (use_wt) {
    dim3 g((Nn + 31) / 32, (C + 31) / 32);
    k_transpose<<<g, dim3(32, 8), 0, stream>>>(W, Wt, Nn);
  }

  // 4) scatter-add over edges
  const int num_tiles = E / TILE_EDGES;
  if (num_tiles > 0) {
    const int grid = num_tiles < SCATTER_GRID ? num_tiles : SCATTER_GRID;
    k_scatter<<<grid, SCATTER_BLOCK, 0, stream>>>(
        rows, cols, use_wt ? Wt : W, out, rmin, num_tiles, Nn, use_wt);
  }
  const int tail0 = num_tiles * TILE_EDGES;
  if (tail0 < E) {
    const int tail_edges = E - tail0;
    const int waves      = (tail_edges + 1) / 2;
    const int blocks     = (waves * 32 + 255) / 256;
    k_scatter_tail<<<blocks, 256, 0, stream>>>(
        rows, cols, use_wt ? Wt : W, out, rmin, tail0, E, Nn, use_wt);
  }
}